// FeatureLoss_85083302134400
// MI455X (gfx1250) — compile-verified
//
#include <hip/hip_runtime.h>
#include <math.h>

// ---------------------------------------------------------------------------
// FGD loss for MI455X (gfx1250).
// Memory-bound: 2 streaming passes over preds_S/preds_T (~500MB total @23.3TB/s
// ~= 21us floor). Tiny MLPs done with V_WMMA_F32_16X16X4_F32 (exact f32 WMMA).
// ---------------------------------------------------------------------------

#define NB   8
#define CC   256
#define C2H  128
#define HH   100
#define WW   152
#define HWS  (HH * WW)          // 15200
#define NTB  20
#define TEMP_   0.5f
#define ALPHA_  0.001f
#define BETA_   0.0005f
#define GAMMA_  0.001f
#define LAMBDA_ 5e-06f
#define LN_EPS_ 1e-05f

typedef float v2f __attribute__((ext_vector_type(2)));
typedef float v8f __attribute__((ext_vector_type(8)));

// workspace layout in floats
enum {
  OFF_ABS_S   = 0,                        // N*HW  : sum_c |S|
  OFF_ABS_T   = OFF_ABS_S + NB * HWS,     // N*HW  : sum_c |T|
  OFF_D2      = OFF_ABS_T + NB * HWS,     // N*HW  : sum_c (S-T)^2
  OFF_CM_S    = OFF_D2 + NB * HWS,        // N*HW  : sum_c cms_w[c]*S
  OFF_CM_T    = OFF_CM_S + NB * HWS,      // N*HW  : sum_c cmt_w[c]*T
  OFF_P_S     = OFF_CM_T + NB * HWS,      // N*HW  : softmax(cm_s)
  OFF_P_T     = OFF_P_S + NB * HWS,       // N*HW  : softmax(cm_t)
  OFF_MFG     = OFF_P_T + NB * HWS,       // N*HW  : Mask_fg
  OFF_BGS     = OFF_MFG + NB * HWS,       // N     : 1/sum(bg) (or 0)
  OFF_SMAX    = OFF_BGS + NB,             // 4*N   : softmax maxes (fea_s, fea_t, cm_s, cm_t)
  OFF_SSUM    = OFF_SMAX + 4 * NB,        // 4*N   : softmax expsums
  OFF_CHS     = OFF_SSUM + 4 * NB,        // N*C   : sum_hw |S|
  OFF_CHT     = OFF_CHS + NB * CC,        // N*C   : sum_hw |T|
  OFF_SUMD    = OFF_CHT + NB * CC,        // N*C   : sum_hw (S-T)
  OFF_SUMD2   = OFF_SUMD + NB * CC,       // N*C   : sum_hw (S-T)^2
  OFF_CTXS    = OFF_SUMD2 + NB * CC,      // N*C   : ctx_s
  OFF_CTXT    = OFF_CTXS + NB * CC,       // N*C   : ctx_t
  OFF_DELS    = OFF_CTXT + NB * CC,       // N*C   : channel_add(ctx_s)
  OFF_DELT    = OFF_DELS + NB * CC,       // N*C   : channel_add(ctx_t)
  OFF_PART_SP = OFF_DELT + NB * CC,       // N     : spatial loss partials (x N*loss)
  OFF_PART_CA = OFF_PART_SP + NB,         // 1     : channel-attn partial
  OFF_PART_RE = OFF_PART_CA + 1,          // N     : rela partials
  WS_FLOATS   = OFF_PART_RE + NB
};

__device__ __forceinline__ float block_reduce_sum(float v, float* red) {
  int tid = threadIdx.x;
  red[tid] = v;
  __syncthreads();
  for (int s = blockDim.x >> 1; s > 0; s >>= 1) {
    if (tid < s) red[tid] += red[tid + s];
    __syncthreads();
  }
  float r = red[0];
  __syncthreads();
  return r;
}

__device__ __forceinline__ float block_reduce_max(float v, float* red) {
  int tid = threadIdx.x;
  red[tid] = v;
  __syncthreads();
  for (int s = blockDim.x >> 1; s > 0; s >>= 1) {
    if (tid < s) red[tid] = fmaxf(red[tid], red[tid + s]);
    __syncthreads();
  }
  float r = red[0];
  __syncthreads();
  return r;
}

// ---------------------------------------------------------------------------
// Pass A: stream S,T once; per-(n,hw) reductions over C. float4 vectorized.
// ---------------------------------------------------------------------------
__global__ void pass_a(const float* __restrict__ S, const float* __restrict__ T,
                       const float* __restrict__ cms_w, const float* __restrict__ cmt_w,
                       float* __restrict__ ws) {
  const int n = blockIdx.y;
  const int hw0 = blockIdx.x * 1024 + threadIdx.x * 4;
  if (hw0 >= HWS) return;  // HW % 4 == 0 so vectors are all-in or all-out
  const float* Sn = S + (size_t)n * CC * HWS;
  const float* Tn = T + (size_t)n * CC * HWS;
  float4 aS = make_float4(0.f, 0.f, 0.f, 0.f);
  float4 aT = make_float4(0.f, 0.f, 0.f, 0.f);
  float4 d2 = make_float4(0.f, 0.f, 0.f, 0.f);
  float4 cs = make_float4(0.f, 0.f, 0.f, 0.f);
  float4 ct = make_float4(0.f, 0.f, 0.f, 0.f);
  for (int c = 0; c < CC; ++c) {
    if (c + 8 < CC) {  // lightweight prefetch -> global_prefetch_b8
      __builtin_prefetch(Sn + (size_t)(c + 8) * HWS + hw0, 0, 0);
      __builtin_prefetch(Tn + (size_t)(c + 8) * HWS + hw0, 0, 0);
    }
    const float4 s4 = *(const float4*)(Sn + (size_t)c * HWS + hw0);
    const float4 t4 = *(const float4*)(Tn + (size_t)c * HWS + hw0);
    const float wsc = cms_w[c];
    const float wtc = cmt_w[c];
    aS.x += fabsf(s4.x); aS.y += fabsf(s4.y); aS.z += fabsf(s4.z); aS.w += fabsf(s4.w);
    aT.x += fabsf(t4.x); aT.y += fabsf(t4.y); aT.z += fabsf(t4.z); aT.w += fabsf(t4.w);
    float dx = s4.x - t4.x, dy = s4.y - t4.y, dz = s4.z - t4.z, dw = s4.w - t4.w;
    d2.x += dx * dx; d2.y += dy * dy; d2.z += dz * dz; d2.w += dw * dw;
    cs.x += wsc * s4.x; cs.y += wsc * s4.y; cs.z += wsc * s4.z; cs.w += wsc * s4.w;
    ct.x += wtc * t4.x; ct.y += wtc * t4.y; ct.z += wtc * t4.z; ct.w += wtc * t4.w;
  }
  const int o = n * HWS + hw0;
  *(float4*)(ws + OFF_ABS_S + o) = aS;
  *(float4*)(ws + OFF_ABS_T + o) = aT;
  *(float4*)(ws + OFF_D2 + o)    = d2;
  *(float4*)(ws + OFF_CM_S + o)  = cs;
  *(float4*)(ws + OFF_CM_T + o)  = ct;
}

// ---------------------------------------------------------------------------
// Foreground/background masks from gt bboxes. One block per sample.
// ---------------------------------------------------------------------------
__global__ void masks_kernel(const float* __restrict__ gt, const int* __restrict__ img_h,
                             const int* __restrict__ img_w, float* __restrict__ ws) {
  __shared__ float shmin[NTB], shmax[NTB], swmin[NTB], swmax[NTB], sarea[NTB];
  __shared__ float red[256];
  const int n = blockIdx.x, tid = threadIdx.x;
  if (tid < NTB) {
    const float* g = gt + (size_t)(n * NTB + tid) * 4;
    const float sw = (float)WW / (float)img_w[0];
    const float sh = (float)HH / (float)img_h[0];
    const float wmn = floorf(g[0] * sw), hmn = floorf(g[1] * sh);
    const float wmx = ceilf(g[2] * sw), hmx = ceilf(g[3] * sh);
    swmin[tid] = wmn; swmax[tid] = wmx; shmin[tid] = hmn; shmax[tid] = hmx;
    sarea[tid] = 1.0f / ((hmx + 1.0f - hmn) * (wmx + 1.0f - wmn));
  }
  __syncthreads();
  float bgcnt = 0.f;
  for (int hw = tid; hw < HWS; hw += blockDim.x) {
    const float h = (float)(hw / WW);
    const float w = (float)(hw % WW);
    float fg = 0.f;
#pragma unroll
    for (int b = 0; b < NTB; ++b) {
      const bool in = (h >= shmin[b]) & (h <= shmax[b]) & (w >= swmin[b]) & (w <= swmax[b]);
      fg = fmaxf(fg, in ? sarea[b] : 0.f);
    }
    ws[OFF_MFG + n * HWS + hw] = fg;
    bgcnt += (fg > 0.f) ? 0.f : 1.f;
  }
  const float s = block_reduce_sum(bgcnt, red);
  if (tid == 0) ws[OFF_BGS + n] = (s > 0.f) ? (1.0f / s) : 0.f;
}

// ---------------------------------------------------------------------------
// Softmax stats (max, expsum) over HW for fea_s, fea_t, cm_s, cm_t.
// grid = (N, 4)
// ---------------------------------------------------------------------------
__global__ void softmax_stats(const float* __restrict__ cms_b, const float* __restrict__ cmt_b,
                              float* __restrict__ ws) {
  const int n = blockIdx.x, kind = blockIdx.y, tid = threadIdx.x;
  __shared__ float red[256];
  const float* src;
  float scale = 1.f, bias = 0.f;
  if (kind == 0)      { src = ws + OFF_ABS_S + n * HWS; scale = 1.f / (CC * TEMP_); }
  else if (kind == 1) { src = ws + OFF_ABS_T + n * HWS; scale = 1.f / (CC * TEMP_); }
  else if (kind == 2) { src = ws + OFF_CM_S + n * HWS;  bias = cms_b[0]; }
  else                { src = ws + OFF_CM_T + n * HWS;  bias = cmt_b[0]; }
  float m = -INFINITY;
  for (int hw = tid; hw < HWS; hw += blockDim.x) m = fmaxf(m, src[hw] * scale + bias);
  m = block_reduce_max(m, red);
  float se = 0.f;
  for (int hw = tid; hw < HWS; hw += blockDim.x) se += expf(src[hw] * scale + bias - m);
  se = block_reduce_sum(se, red);
  if (tid == 0) {
    ws[OFF_SMAX + kind * NB + n] = m;
    ws[OFF_SSUM + kind * NB + n] = se;
  }
}

// ---------------------------------------------------------------------------
// Per-(n,hw) finalize: spatial attn L1 + fg/bg losses; write softmax probs p.
// One block per sample. Partial = contribution to N*loss.
// ---------------------------------------------------------------------------
__global__ void spatial_finalize(const float* __restrict__ cms_b, const float* __restrict__ cmt_b,
                                 float* __restrict__ ws) {
  const int n = blockIdx.x, tid = threadIdx.x;
  __shared__ float red[256];
  const float m0 = ws[OFF_SMAX + 0 * NB + n], s0 = ws[OFF_SSUM + 0 * NB + n];
  const float m1 = ws[OFF_SMAX + 1 * NB + n], s1 = ws[OFF_SSUM + 1 * NB + n];
  const float m2 = ws[OFF_SMAX + 2 * NB + n], s2 = ws[OFF_SSUM + 2 * NB + n];
  const float m3 = ws[OFF_SMAX + 3 * NB + n], s3 = ws[OFF_SSUM + 3 * NB + n];
  const float bs = ws[OFF_BGS + n];
  const float b2 = cms_b[0], b3 = cmt_b[0];
  const float fsc = 1.f / (CC * TEMP_);
  float acc = 0.f;
  for (int hw = tid; hw < HWS; hw += blockDim.x) {
    const int o = n * HWS + hw;
    const float Ss = expf(ws[OFF_ABS_S + o] * fsc - m0) * ((float)HWS / s0);
    const float St = expf(ws[OFF_ABS_T + o] * fsc - m1) * ((float)HWS / s1);
    acc += GAMMA_ * fabsf(Ss - St);
    const float fg = ws[OFF_MFG + o];
    const float bg = (fg > 0.f) ? 0.f : bs;
    acc += (ALPHA_ * fg * fg + BETA_ * bg * bg) * ws[OFF_D2 + o];
    ws[OFF_P_S + o] = expf(ws[OFF_CM_S + o] + b2 - m2) / s2;
    ws[OFF_P_T + o] = expf(ws[OFF_CM_T + o] + b3 - m3) / s3;
  }
  acc = block_reduce_sum(acc, red);
  if (tid == 0) ws[OFF_PART_SP + n] = acc;
}

// ---------------------------------------------------------------------------
// Pass B: stream S,T again; per-(n,c) reductions over HW (ch, sum d, sum d^2,
// ctx = sum x*p). grid = (C, N), one block per (n,c) row.
// ---------------------------------------------------------------------------
__global__ void pass_b(const float* __restrict__ S, const float* __restrict__ T,
                       float* __restrict__ ws) {
  const int c = blockIdx.x, n = blockIdx.y, tid = threadIdx.x;
  __shared__ float red[256];
  const float* Sp = S + ((size_t)n * CC + c) * HWS;
  const float* Tp = T + ((size_t)n * CC + c) * HWS;
  const float* ps = ws + OFF_P_S + n * HWS;
  const float* pt = ws + OFF_P_T + n * HWS;
  float aS = 0.f, aT = 0.f, ad = 0.f, ad2 = 0.f, cxs = 0.f, cxt = 0.f;
  for (int hw = tid * 4; hw < HWS; hw += 1024) {
    const float4 s4 = *(const float4*)(Sp + hw);
    const float4 t4 = *(const float4*)(Tp + hw);
    const float4 p4 = *(const float4*)(ps + hw);
    const float4 q4 = *(const float4*)(pt + hw);
    aS += fabsf(s4.x) + fabsf(s4.y) + fabsf(s4.z) + fabsf(s4.w);
    aT += fabsf(t4.x) + fabsf(t4.y) + fabsf(t4.z) + fabsf(t4.w);
    const float dx = s4.x - t4.x, dy = s4.y - t4.y, dz = s4.z - t4.z, dw = s4.w - t4.w;
    ad += dx + dy + dz + dw;
    ad2 += dx * dx + dy * dy + dz * dz + dw * dw;
    cxs += s4.x * p4.x + s4.y * p4.y + s4.z * p4.z + s4.w * p4.w;
    cxt += t4.x * q4.x + t4.y * q4.y + t4.z * q4.z + t4.w * q4.w;
  }
  aS  = block_reduce_sum(aS, red);
  aT  = block_reduce_sum(aT, red);
  ad  = block_reduce_sum(ad, red);
  ad2 = block_reduce_sum(ad2, red);
  cxs = block_reduce_sum(cxs, red);
  cxt = block_reduce_sum(cxt, red);
  if (tid == 0) {
    const int o = n * CC + c;
    ws[OFF_CHS + o]   = aS;
    ws[OFF_CHT + o]   = aT;
    ws[OFF_SUMD + o]  = ad;
    ws[OFF_SUMD2 + o] = ad2;
    ws[OFF_CTXS + o]  = cxs;
    ws[OFF_CTXT + o]  = cxt;
  }
}

// ---------------------------------------------------------------------------
// Channel attention L1: GAMMA * sum |C*softmax(ch_s/T) - C*softmax(ch_t/T)|
// ---------------------------------------------------------------------------
__global__ void channel_att(float* __restrict__ ws) {
  const int tid = threadIdx.x;  // 256 threads == C
  __shared__ float red[256];
  const float sc = 1.f / ((float)HWS * TEMP_);
  float acc = 0.f;
  for (int n = 0; n < NB; ++n) {
    const float vs = ws[OFF_CHS + n * CC + tid] * sc;
    const float vt = ws[OFF_CHT + n * CC + tid] * sc;
    const float ms = block_reduce_max(vs, red);
    const float ss = block_reduce_sum(expf(vs - ms), red);
    const float mt = block_reduce_max(vt, red);
    const float st = block_reduce_sum(expf(vt - mt), red);
    const float Cs = (float)CC * expf(vs - ms) / ss;
    const float Ct = (float)CC * expf(vt - mt) / st;
    acc += GAMMA_ * fabsf(Cs - Ct);
  }
  acc = block_reduce_sum(acc, red);
  if (tid == 0) ws[OFF_PART_CA] = acc;
}

// ---------------------------------------------------------------------------
// channel_add MLP via V_WMMA_F32_16X16X4_F32 (exact f32 path).
// grid.x = 2 (student / teacher), 256 threads = 8 wave32.
// Stage 1: y1 = ctx @ w1^T + b1  (M=16 padded, N=128, K=256)
// Stage 2: LayerNorm + ReLU      (rows 0..7)
// Stage 3: delta = act @ w2^T + b2 (M=16 padded, N=256, K=128)
// ---------------------------------------------------------------------------
__global__ void mlp_wmma(const float* __restrict__ w1s, const float* __restrict__ b1s,
                         const float* __restrict__ lnws, const float* __restrict__ lnbs,
                         const float* __restrict__ w2s, const float* __restrict__ b2s,
                         const float* __restrict__ w1t, const float* __restrict__ b1t,
                         const float* __restrict__ lnwt, const float* __restrict__ lnbt,
                         const float* __restrict__ w2t, const float* __restrict__ b2t,
                         float* __restrict__ ws) {
  const bool isT = (blockIdx.x == 1);
  const float* w1  = isT ? w1t : w1s;
  const float* b1  = isT ? b1t : b1s;
  const float* lnw = isT ? lnwt : lnws;
  const float* lnb = isT ? lnbt : lnbs;
  const float* w2  = isT ? w2t : w2s;
  const float* b2  = isT ? b2t : b2s;
  const float* ctx = ws + (isT ? OFF_CTXT : OFF_CTXS);
  float* delta     = ws + (isT ? OFF_DELT : OFF_DELS);

  __shared__ float A0[16 * CC];   // padded ctx (rows 8..15 zero)
  __shared__ float Y1[16 * C2H];  // y1 / activations (rows 8..15 zero)
  const int tid = threadIdx.x;
  for (int i = tid; i < 16 * CC; i += 256) A0[i] = 0.f;
  for (int i = tid; i < 16 * C2H; i += 256) Y1[i] = 0.f;
  __syncthreads();
  for (int i = tid; i < NB * CC; i += 256) A0[i] = ctx[i];
  __syncthreads();

  const int wv = tid >> 5;
  const int lane = tid & 31;
  const int row = lane & 15;
  const int koff = (lane < 16) ? 0 : 2;  // ISA: lanes 16..31 hold K=2,3 (f32 A/B 16x4)

  // ---- Stage 1: 8 waves x one 16x16 j-tile each, K = 256 ----
  {
    const int j = wv * 16 + row;
    v8f acc = {0.f, 0.f, 0.f, 0.f, 0.f, 0.f, 0.f, 0.f};
    for (int k0 = 0; k0 < CC; k0 += 4) {
      v2f a, b;
      a.x = A0[row * CC + k0 + koff];
      a.y = A0[row * CC + k0 + koff + 1];
      b.x = w1[j * CC + k0 + koff];
      b.y = w1[j * CC + k0 + koff + 1];
      acc = __builtin_amdgcn_wmma_f32_16x16x4_f32(false, a, false, b, (short)0, acc,
                                                  false, false);
    }
    if (lane < 16) {
#pragma unroll
      for (int r = 0; r < 8; ++r)  // rows 0..7 valid (n index), rows 8..15 are pad
        Y1[r * C2H + j] = acc[r] + b1[j];
    }
  }
  __syncthreads();

  // ---- Stage 2: LayerNorm + ReLU, one thread per sample row ----
  if (tid < NB) {
    float mu = 0.f;
    for (int j = 0; j < C2H; ++j) mu += Y1[tid * C2H + j];
    mu *= (1.f / C2H);
    float var = 0.f;
    for (int j = 0; j < C2H; ++j) {
      const float d = Y1[tid * C2H + j] - mu;
      var += d * d;
    }
    var *= (1.f / C2H);
    const float inv = rsqrtf(var + LN_EPS_);
    for (int j = 0; j < C2H; ++j) {
      const float y = (Y1[tid * C2H + j] - mu) * inv * lnw[j] + lnb[j];
      Y1[tid * C2H + j] = fmaxf(y, 0.f);
    }
  }
  __syncthreads();

  // ---- Stage 3: 16 c-tiles, 2 per wave, K = 128 ----
  for (int t2 = 0; t2 < 2; ++t2) {
    const int cidx = (wv + t2 * 8) * 16 + row;
    v8f acc = {0.f, 0.f, 0.f, 0.f, 0.f, 0.f, 0.f, 0.f};
    for (int k0 = 0; k0 < C2H; k0 += 4) {
      v2f a, b;
      a.x = Y1[row * C2H + k0 + koff];
      a.y = Y1[row * C2H + k0 + koff + 1];
      b.x = w2[cidx * C2H + k0 + koff];
      b.y = w2[cidx * C2H + k0 + koff + 1];
      acc = __builtin_amdgcn_wmma_f32_16x16x4_f32(false, a, false, b, (short)0, acc,
                                                  false, false);
    }
    if (lane < 16) {
#pragma unroll
      for (int r = 0; r < 8; ++r)
        delta[r * CC + cidx] = acc[r] + b2[cidx];
    }
  }
}

// ---------------------------------------------------------------------------
// rela_loss partials: sum_c [ sumd2 + 2*e*sumd + HW*e^2 ] * LAMBDA per sample.
// ---------------------------------------------------------------------------
__global__ void rela_kernel(float* __restrict__ ws) {
  const int n = blockIdx.x, c = threadIdx.x;
  __shared__ float red[256];
  const int o = n * CC + c;
  const float e = ws[OFF_DELS + o] - ws[OFF_DELT + o];
  float val = LAMBDA_ * (ws[OFF_SUMD2 + o] + 2.f * e * ws[OFF_SUMD + o] + (float)HWS * e * e);
  val = block_reduce_sum(val, red);
  if (c == 0) ws[OFF_PART_RE + n] = val;
}

// Deterministic fixed-order final sum -> d_out[0] = (N*loss)/N
__global__ void final_kernel(const float* __restrict__ ws, float* __restrict__ out) {
  if (threadIdx.x == 0 && blockIdx.x == 0) {
    float t = 0.f;
    for (int i = 0; i < NB; ++i) t += ws[OFF_PART_SP + i];
    t += ws[OFF_PART_CA];
    for (int i = 0; i < NB; ++i) t += ws[OFF_PART_RE + i];
    out[0] = t / (float)NB;
  }
}

extern "C" void kernel_launch(void* const* d_in, const int* in_sizes, int n_in,
                              void* d_out, int out_size, void* d_ws, size_t ws_size,
                              hipStream_t stream) {
  const float* S      = (const float*)d_in[0];
  const float* T      = (const float*)d_in[1];
  const float* gt     = (const float*)d_in[2];
  const float* cms_w  = (const float*)d_in[3];
  const float* cms_b  = (const float*)d_in[4];
  const float* cmt_w  = (const float*)d_in[5];
  const float* cmt_b  = (const float*)d_in[6];
  const float* cas_w1 = (const float*)d_in[7];
  const float* cas_b1 = (const float*)d_in[8];
  const float* cas_lnw = (const float*)d_in[9];
  const float* cas_lnb = (const float*)d_in[10];
  const float* cas_w2 = (const float*)d_in[11];
  const float* cas_b2 = (const float*)d_in[12];
  const float* cat_w1 = (const float*)d_in[13];
  const float* cat_b1 = (const float*)d_in[14];
  const float* cat_lnw = (const float*)d_in[15];
  const float* cat_lnb = (const float*)d_in[16];
  const float* cat_w2 = (const float*)d_in[17];
  const float* cat_b2 = (const float*)d_in[18];
  const int* img_h    = (const int*)d_in[19];
  const int* img_w    = (const int*)d_in[20];
  float* ws = (float*)d_ws;

  dim3 gA((HWS + 1023) / 1024, NB);
  pass_a<<<gA, 256, 0, stream>>>(S, T, cms_w, cmt_w, ws);
  masks_kernel<<<NB, 256, 0, stream>>>(gt, img_h, img_w, ws);
  softmax_stats<<<dim3(NB, 4), 256, 0, stream>>>(cms_b, cmt_b, ws);
  spatial_finalize<<<NB, 256, 0, stream>>>(cms_b, cmt_b, ws);
  pass_b<<<dim3(CC, NB), 256, 0, stream>>>(S, T, ws);
  channel_att<<<1, 256, 0, stream>>>(ws);
  mlp_wmma<<<2, 256, 0, stream>>>(cas_w1, cas_b1, cas_lnw, cas_lnb, cas_w2, cas_b2,
                                  cat_w1, cat_b1, cat_lnw, cat_lnb, cat_w2, cat_b2, ws);
  rela_kernel<<<NB, 256, 0, stream>>>(ws);
  final_kernel<<<1, 64, 0, stream>>>(ws, (float*)d_out);
}